// OnnxPostprocess_38723425140757
// MI455X (gfx1250) — compile-verified
//
#include <hip/hip_runtime.h>
#include <hip/hip_bf16.h>
#include <stdint.h>

#define NUM_ANCHORS 3
#define NUM_CLASSES 80
#define HW_CELLS 1444
#define NBOX (NUM_ANCHORS * HW_CELLS)   // 4332
#define NPAD 8192                        // next pow2 for bitonic sort
#define BATCH 4
#define SCORE_THRESH 0.5f
#define NMS_THRESH 0.45f
#define THREADS 256                      // 8 wave32 waves
#define SLOTS ((NBOX + THREADS - 1) / THREADS)   // 17 j-slots per thread

// One workgroup per (batch, class) problem. Entire problem lives in LDS
// (CDNA5: 320KB per WGP workgroup): 64KB sort keys + 69KB sorted boxes + flags.
// NMS inner sweep is register-resident: each thread holds its 17 boxes+areas
// in VGPRs and a live-bit mask, so per-step LDS traffic is just the pivot
// broadcast plus rare suppression-flag writes.
__global__ __launch_bounds__(THREADS, 1)
void OnnxPostprocess_nms_kernel(const float* __restrict__ boxes,
                                const float* __restrict__ scores,
                                float* __restrict__ out)
{
    __shared__ unsigned long long keys[NPAD];   // [orderable_f32_key:32 | ~index:32]
    __shared__ float4 boxS[NBOX];               // boxes gathered in sorted order
    __shared__ unsigned char keepS[NBOX];       // greedy-NMS keep flags

    const int tid = threadIdx.x;
    const int b   = blockIdx.x / NUM_CLASSES;
    const int c   = blockIdx.x % NUM_CLASSES;

    // ---- Phase 1: load class-c scores, threshold, pack descending-sort keys ----
    // key32 is a monotone map of float: sort(key32 desc) == sort(prob desc).
    // Low 32 bits hold ~index so equal probs order by ascending index under a
    // plain descending u64 sort (matches argsort stability).
    const float* sbase = scores + (size_t)b * NBOX * NUM_CLASSES + c;
    for (int n = tid; n < NPAD; n += THREADS) {
        unsigned long long kv = 0ull;           // pads sort to the very end
        if (n < NBOX) {
            float s = sbase[(size_t)n * NUM_CLASSES];
            float p = (s > SCORE_THRESH) ? s : -1.0f;
            unsigned u   = __float_as_uint(p);
            unsigned k32 = (u & 0x80000000u) ? ~u : (u | 0x80000000u);
            kv = ((unsigned long long)k32 << 32) | (unsigned)(~n);
        }
        keys[n] = kv;
    }
    __syncthreads();

    // ---- Phase 2: in-LDS bitonic sort, descending on u64 key ----
    for (int k = 2; k <= NPAD; k <<= 1) {
        for (int j = k >> 1; j > 0; j >>= 1) {
            for (int s = 0; s < NPAD / THREADS; ++s) {
                int i   = tid + s * THREADS;
                int ixj = i ^ j;
                if (ixj > i) {
                    unsigned long long a = keys[i];
                    unsigned long long q = keys[ixj];
                    bool sw = ((i & k) == 0) ? (a < q) : (a > q);
                    if (sw) { keys[i] = q; keys[ixj] = a; }
                }
            }
            __syncthreads();
        }
    }

    // ---- Phase 3: gather boxes in sorted order via async global->LDS (ASYNCcnt) ----
    // Per-lane indexed gather: 16B box goes straight into LDS, no VGPR bounce.
    const float* bbase = boxes + (size_t)b * NBOX * 4;
    for (int jj = tid; jj < NBOX; jj += THREADS) {
        unsigned long long kv = keys[jj];
        unsigned k32 = (unsigned)(kv >> 32);
        int n = (int)(~(unsigned)kv);                        // original box index
        keepS[jj] = (k32 > 0x80000000u) ? 1 : 0;             // prob > 0 (above thresh)
        unsigned ldsAddr = (unsigned)(uintptr_t)(&boxS[jj]); // low 32 bits = LDS offset
        unsigned long long gaddr =
            (unsigned long long)(uintptr_t)(bbase + (size_t)n * 4);
        asm volatile("global_load_async_to_lds_b128 %0, %1, off"
                     :: "v"(ldsAddr), "v"(gaddr)
                     : "memory");
    }
    asm volatile("s_wait_asynccnt 0x0" ::: "memory");
    __syncthreads();

    // ---- Phase 3.5: pull owned slots into registers ----
    float4 bx[SLOTS];
    float  ar[SLOTS];
    unsigned live = 0u;                  // bit s: slot jj = tid + s*THREADS alive
    for (int s = 0; s < SLOTS; ++s) {
        int jj = tid + s * THREADS;
        if (jj < NBOX) {
            float4 v = boxS[jj];
            bx[s] = v;
            ar[s] = (v.z - v.x) * (v.w - v.y);
            if (keepS[jj]) live |= (1u << s);
        }
    }

    // ---- Phase 4: greedy NMS (sequential pivots; register-resident sweep) ----
    for (int i = 0; i < NBOX; ++i) {
        __syncthreads();                 // make iteration i-1's suppressions visible
        if (!keepS[i]) continue;         // uniform across block (same LDS read)
        float4 bi = boxS[i];             // LDS broadcast
        float areai = (bi.z - bi.x) * (bi.w - bi.y);
        // first slot with jj = tid + s*THREADS > i
        int s0 = (i >= tid) ? (((i - tid) >> 8) + 1) : 0;
        for (int s = s0; s < SLOTS; ++s) {
            if (!(live & (1u << s))) continue;
            float4 bj  = bx[s];
            float tlx  = fmaxf(bi.x, bj.x);
            float tly  = fmaxf(bi.y, bj.y);
            float brx  = fminf(bi.z, bj.z);
            float bry  = fminf(bi.w, bj.w);
            float w    = fmaxf(brx - tlx, 0.0f);
            float h    = fmaxf(bry - tly, 0.0f);
            float inter = w * h;
            float iou   = inter / (areai + ar[s] - inter);
            if (iou > NMS_THRESH) {
                live &= ~(1u << s);
                keepS[tid + s * THREADS] = 0;   // publish for future pivot reads
            }
        }
    }
    __syncthreads();

    // ---- Phase 5: emit padded [N,6] detections (zeros for suppressed rows) ----
    float* obase = out + (size_t)(b * NUM_CLASSES + c) * NBOX * 6;
    for (int s = 0; s < SLOTS; ++s) {
        int jj = tid + s * THREADS;
        if (jj >= NBOX) break;
        float* o = obase + (size_t)jj * 6;
        if (live & (1u << s)) {
            unsigned long long kv = keys[jj];
            unsigned k32 = (unsigned)(kv >> 32);
            unsigned u   = (k32 & 0x80000000u) ? (k32 & 0x7FFFFFFFu) : ~k32;
            float p = __uint_as_float(u);
            float4 v = bx[s];
            o[0] = v.x; o[1] = v.y; o[2] = v.z; o[3] = v.w;
            o[4] = p;   o[5] = (float)c;
        } else {
            o[0] = 0.0f; o[1] = 0.0f; o[2] = 0.0f;
            o[3] = 0.0f; o[4] = 0.0f; o[5] = 0.0f;
        }
    }
}

extern "C" void kernel_launch(void* const* d_in, const int* in_sizes, int n_in,
                              void* d_out, int out_size, void* d_ws, size_t ws_size,
                              hipStream_t stream) {
    const float* boxes  = (const float*)d_in[0];   // [B, A, HW, 4] = [B, NBOX, 4]
    const float* scores = (const float*)d_in[1];   // [B, A, HW, C] = [B, NBOX, C]
    float* out = (float*)d_out;                    // [B, C, NBOX, 6]
    (void)in_sizes; (void)n_in; (void)out_size; (void)d_ws; (void)ws_size;
    dim3 grid(BATCH * NUM_CLASSES);                // 320 independent (b,c) problems
    dim3 block(THREADS);
    OnnxPostprocess_nms_kernel<<<grid, block, 0, stream>>>(boxes, scores, out);
}